// DJVerifier_87058987090549
// MI455X (gfx1250) — compile-verified
//
#include <hip/hip_runtime.h>
#include <math.h>

// ---------------------------------------------------------------------------
// Reference:  p = (||tm1-vmask1|| + ||tm2-vmask2||) / 38400
//             med_k = exact lower median (sorted[(n-1)//2])
//             q = (||(tm1>=m1)-amask1|| + ||(tm2>=m2)-amask2||) / 384
// tm1 = c2[:,:,7,7]  (100,128)  elem stride 56*56=3136, offset 7*56+7=399
// tm2 = c3[:,:,3,3]  (100,256)  elem stride 28*28= 784, offset 3*28+3= 87
// ---------------------------------------------------------------------------

typedef __attribute__((ext_vector_type(2))) float v2f;
typedef __attribute__((ext_vector_type(8))) float v8f;

#define N1 12800
#define N2 25600
// workspace (float indices)
#define WS_V1   0          // gathered tm1
#define WS_V2   12800      // gathered tm2
#define WS_SCAL 38400      // [2]=t1 [3]=t2 [4]=med1 [5]=med2
#define WS_PART (WS_SCAL + 8) // 48 per-block p-partials
#define K1_BLOCKS_S1 16
#define K1_BLOCKS    48

__device__ __forceinline__ unsigned f2key(float f) {
  unsigned u = __float_as_uint(f);
  return (u & 0x80000000u) ? ~u : (u | 0x80000000u);  // monotone total order
}
__device__ __forceinline__ float key2f(unsigned k) {
  unsigned u = (k & 0x80000000u) ? (k ^ 0x80000000u) : ~k;
  return __uint_as_float(u);
}

// Wave32 sum via V_WMMA_F32_16X16X4_F32 with B = ones:
// lane m holds A[m%16][0 or 2] = v (A[.][1,3] = 0).  D[m][n] = rowsum(m) =
// v[lane m] + v[lane m+16].  C/D layout: lane 0's 8 VGPRs = rows 0..7,
// lane 16's = rows 8..15, so (sum8 @ lane0) + (sum8 @ lane16) = wave total.
// EXEC must be all ones: called only at convergent points of full waves.
__device__ __forceinline__ void waveSumWMMA(float v, float* out2, int lid) {
  v2f a; a.x = v;    a.y = 0.0f;
  v2f b; b.x = 1.0f; b.y = 1.0f;
  v8f c = {};
  v8f d = __builtin_amdgcn_wmma_f32_16x16x4_f32(
      /*neg_a=*/false, a, /*neg_b=*/false, b,
      /*c_mod=*/(short)0, c, /*reuse_a=*/false, /*reuse_b=*/false);
  float s = d[0] + d[1] + d[2] + d[3] + d[4] + d[5] + d[6] + d[7];
  if (lid == 0)  out2[0] = s;   // rows 0..7
  if (lid == 16) out2[1] = s;   // rows 8..15
}

// K1: strided gather into workspace + fused (tm - vmask)^2 partial sums.
__global__ __launch_bounds__(256) void k_gather_sumsq(
    const float* __restrict__ c2, const float* __restrict__ c3,
    const float* __restrict__ vm1, const float* __restrict__ vm2,
    float* __restrict__ ws) {
  __shared__ float wsum[16];  // 8 waves * 2 halves
  const int bid = blockIdx.x;
  const float* src; const float* vm; float* dst;
  long stride, off; int n, b0, nb;
  if (bid < K1_BLOCKS_S1) { src = c2; vm = vm1; dst = ws + WS_V1;
    stride = 3136; off = 399; n = N1; b0 = bid; nb = K1_BLOCKS_S1; }
  else { src = c3; vm = vm2; dst = ws + WS_V2;
    stride = 784;  off = 87;  n = N2; b0 = bid - K1_BLOCKS_S1;
    nb = K1_BLOCKS - K1_BLOCKS_S1; }
  const int tid = threadIdx.x;
  const int step = nb * 256;
  float acc = 0.0f;
  for (int j = b0 * 256 + tid; j < n; j += step) {
    if (j + step < n)  // prefetch next strided line -> global_prefetch_b8
      __builtin_prefetch(&src[(long)(j + step) * stride + off], 0, 0);
    float v = src[(long)j * stride + off];
    dst[j] = v;
    float dlt = v - vm[j];
    acc = fmaf(dlt, dlt, acc);
  }
  const int lid = tid & 31, wid = tid >> 5;
  waveSumWMMA(acc, &wsum[wid * 2], lid);
  __syncthreads();
  if (tid == 0) {
    float s = 0.0f;
    for (int i = 0; i < 16; ++i) s += wsum[i];
    ws[WS_PART + bid] = s;
  }
}

// K2: one block per set: exact k-th order statistic via 4-pass MSB radix
// select (k = (n-1)/2, matching sorted[(n-1)//2]), then ((v<med)?0:1 - amask)^2.
__global__ __launch_bounds__(1024) void k_median_q(
    const float* __restrict__ am1, const float* __restrict__ am2,
    float* __restrict__ ws) {
  __shared__ unsigned hist[256];
  __shared__ unsigned sh_prefix;
  __shared__ int sh_krem;
  __shared__ float sh_med;
  __shared__ float wsum[64];  // 32 waves * 2 halves
  const int set = blockIdx.x;
  const float* vals = ws + (set ? WS_V2 : WS_V1);
  const float* am   = set ? am2 : am1;
  const int n = set ? N2 : N1;
  const int tid = threadIdx.x;

  if (tid == 0) { sh_prefix = 0u; sh_krem = (n - 1) >> 1; }
  __syncthreads();
  for (int pass = 0; pass < 4; ++pass) {
    const int shift = 24 - 8 * pass;
    const unsigned himask = (pass == 0) ? 0u : (0xFFFFFFFFu << (shift + 8));
    if (tid < 256) hist[tid] = 0u;
    __syncthreads();
    const unsigned pfx = sh_prefix;
    for (int i = tid; i < n; i += 1024) {
      unsigned key = f2key(vals[i]);
      if ((key & himask) == pfx)
        atomicAdd(&hist[(key >> shift) & 0xFFu], 1u);
    }
    __syncthreads();
    if (tid == 0) {
      int kr = sh_krem;
      unsigned p = sh_prefix;
      for (int b = 0; b < 256; ++b) {
        int h = (int)hist[b];
        if (kr < h) { p |= ((unsigned)b) << shift; break; }
        kr -= h;
      }
      sh_prefix = p; sh_krem = kr;
    }
    __syncthreads();
  }
  if (tid == 0) { sh_med = key2f(sh_prefix); ws[WS_SCAL + 4 + set] = sh_med; }
  __syncthreads();

  const float med = sh_med;
  float acc = 0.0f;
  for (int i = tid; i < n; i += 1024) {
    float b = (vals[i] < med) ? 0.0f : 1.0f;
    float dlt = b - am[i];
    acc = fmaf(dlt, dlt, acc);
  }
  const int lid = tid & 31, wid = tid >> 5;
  waveSumWMMA(acc, &wsum[wid * 2], lid);
  __syncthreads();
  if (tid == 0) {
    float s = 0.0f;
    for (int i = 0; i < 64; ++i) s += wsum[i];
    ws[WS_SCAL + 2 + set] = s;  // t1 / t2
  }
}

// K3: deterministic sequential combine of the handful of scalars.
__global__ void k_final(const float* __restrict__ ws, float* __restrict__ out) {
  float s1 = 0.0f, s2 = 0.0f;
  for (int i = 0; i < K1_BLOCKS_S1; ++i) s1 += ws[WS_PART + i];
  for (int i = K1_BLOCKS_S1; i < K1_BLOCKS; ++i) s2 += ws[WS_PART + i];
  float t1 = ws[WS_SCAL + 2], t2 = ws[WS_SCAL + 3];
  out[0] = (sqrtf(s1) + sqrtf(s2)) / 38400.0f;
  out[1] = (sqrtf(t1) + sqrtf(t2)) / 384.0f;
}

extern "C" void kernel_launch(void* const* d_in, const int* in_sizes, int n_in,
                              void* d_out, int out_size, void* d_ws, size_t ws_size,
                              hipStream_t stream) {
  const float* c2  = (const float*)d_in[0];
  const float* c3  = (const float*)d_in[1];
  const float* vm1 = (const float*)d_in[2];
  const float* vm2 = (const float*)d_in[3];
  const float* am1 = (const float*)d_in[4];
  const float* am2 = (const float*)d_in[5];
  float* ws  = (float*)d_ws;
  float* out = (float*)d_out;

  k_gather_sumsq<<<K1_BLOCKS, 256, 0, stream>>>(c2, c3, vm1, vm2, ws);
  k_median_q<<<2, 1024, 0, stream>>>(am1, am2, ws);
  k_final<<<1, 1, 0, stream>>>(ws, out);
}